// UltraLightweightMemory_76441827934741
// MI455X (gfx1250) — compile-verified
//
#include <hip/hip_runtime.h>

typedef __attribute__((ext_vector_type(2))) float v2f;
typedef __attribute__((ext_vector_type(8))) float v8f;

constexpr int B = 32, S = 4096, D = 160, M = 16, R = 16;
constexpr int CH = 32;            // chunks over S for the interface reduction
constexpr int ROWS = S / CH;      // 128 rows per chunk
constexpr int WPAD = 164;         // D=160 padded -> conflict-free lane stride (gcd(164,64)=4, cycle 16)
constexpr int P20 = 20;           // 16 padded -> conflict-free lane stride (gcd(20,64)=4, cycle 16)

// ---------------------------------------------------------------------------
// Kernel 1: partial sums of interface over S (coalesced, deterministic).
// ---------------------------------------------------------------------------
__global__ __launch_bounds__(D)
void reduce_kernel(const float* __restrict__ itf, float* __restrict__ partial)
{
    const int b = blockIdx.x / CH;
    const int c = blockIdx.x % CH;
    const int d = threadIdx.x;
    const float* p = itf + ((size_t)b * S + (size_t)c * ROWS) * D + d;
    float acc = 0.f;
    for (int i = 0; i < ROWS; ++i) acc += p[(size_t)i * D];
    partial[((size_t)b * CH + c) * D + d] = acc;
}

// ---------------------------------------------------------------------------
// Kernel 2: per-batch write path + V_eff precompute.
//   i_mean -> i_proj -> erase/add -> new_memory = LN(memory*(1-erase)+add)
//   V_eff[m][d] = sum_r memory[b,m,r] * W_read_value[d,r]
// ---------------------------------------------------------------------------
__global__ __launch_bounds__(D)
void update_kernel(const float* __restrict__ partial,
                   const float* __restrict__ memory,
                   const float* __restrict__ W_write_key,   // (R,D)
                   const float* __restrict__ W_write_erase, // (D,R)
                   const float* __restrict__ W_write_add,   // (D,R)
                   const float* __restrict__ W_read_value,  // (D,R)
                   float* __restrict__ veff,                // (B,M,D)
                   float* __restrict__ new_mem)             // (B,M,D)
{
    __shared__ float sIm[D];
    __shared__ float sIp[R];
    __shared__ float sMm[M * R];
    __shared__ float sRed[D], sRed2[D];
    __shared__ float sMu, sRstd;

    const int b = blockIdx.x;
    const int d = threadIdx.x;

    float acc = 0.f;
    for (int c = 0; c < CH; ++c) acc += partial[((size_t)b * CH + c) * D + d];
    sIm[d] = acc / (float)S;
    for (int i = d; i < M * R; i += D)
        sMm[i] = memory[((size_t)b * M + (i >> 4)) * D + (i & 15)];
    __syncthreads();

    if (d < R) {
        float s = 0.f;
        for (int j = 0; j < D; ++j) s += sIm[j] * W_write_key[d * D + j];
        sIp[d] = s;
    }
    __syncthreads();

    float er = 0.f, ad = 0.f;
    for (int r = 0; r < R; ++r) {
        er += sIp[r] * W_write_erase[d * R + r];
        ad += sIp[r] * W_write_add[d * R + r];
    }
    er = 1.0f / (1.0f + __expf(-er));

    for (int m = 0; m < M; ++m) {
        float s = 0.f;
        for (int r = 0; r < R; ++r) s += sMm[m * R + r] * W_read_value[d * R + r];
        veff[((size_t)b * M + m) * D + d] = s;
    }

    for (int m = 0; m < M; ++m) {
        float x = memory[((size_t)b * M + m) * D + d] * (1.0f - er) + ad;
        sRed[d] = x; sRed2[d] = x * x;
        __syncthreads();
        if (d < 80) { sRed[d] += sRed[d + 80]; sRed2[d] += sRed2[d + 80]; } __syncthreads();
        if (d < 40) { sRed[d] += sRed[d + 40]; sRed2[d] += sRed2[d + 40]; } __syncthreads();
        if (d < 20) { sRed[d] += sRed[d + 20]; sRed2[d] += sRed2[d + 20]; } __syncthreads();
        if (d < 10) { sRed[d] += sRed[d + 10]; sRed2[d] += sRed2[d + 10]; } __syncthreads();
        if (d < 5)  { sRed[d] += sRed[d + 5];  sRed2[d] += sRed2[d + 5];  } __syncthreads();
        if (d == 0) {
            float s1 = sRed[0] + sRed[1] + sRed[2] + sRed[3] + sRed[4];
            float s2 = sRed2[0] + sRed2[1] + sRed2[2] + sRed2[3] + sRed2[4];
            float mu = s1 / (float)D;
            float var = s2 / (float)D - mu * mu;
            sMu = mu; sRstd = rsqrtf(var + 1e-5f);
        }
        __syncthreads();
        new_mem[((size_t)b * M + m) * D + d] = (x - sMu) * sRstd;
        __syncthreads();
    }
}

// ---------------------------------------------------------------------------
// Kernel 3: fused attention read path, WMMA f32 16x16x4.
// Each wave handles a 16-row S-tile; 8 waves/block share one batch b.
// ---------------------------------------------------------------------------
__global__ __launch_bounds__(256)
void attn_kernel(const float* __restrict__ query,
                 const float* __restrict__ memory,
                 const float* __restrict__ W_read_key,   // (R,D)
                 const float* __restrict__ veff,         // (B,M,D)
                 float* __restrict__ read_out)           // (B,S,D)
{
    __shared__ float sW[16 * WPAD];    // W_read_key[n][k], padded rows
    __shared__ float sMem[16 * P20];   // mem_proj[m][r]
    __shared__ float sVt[D * P20];     // V_eff^T: [d][m]
    __shared__ float sQp[8][16 * P20]; // per-wave q_proj tile (s x r)
    __shared__ float sAt[8][16 * P20]; // per-wave attn tile (s x m)

    const int b    = blockIdx.x >> 5;
    const int tile = blockIdx.x & 31;
    const int w    = threadIdx.x >> 5;
    const int lane = threadIdx.x & 31;
    const int s0   = tile * 128 + w * 16;
    const int t    = threadIdx.x;

    for (int i = t; i < 16 * D; i += 256) {
        int r = i / D, c = i % D;
        sW[r * WPAD + c] = W_read_key[r * D + c];
    }
    for (int i = t; i < 16 * 16; i += 256) {
        int m = i >> 4, r = i & 15;
        sMem[m * P20 + r] = memory[((size_t)b * M + m) * D + r];
    }
    for (int i = t; i < 16 * D; i += 256) {
        int d = i >> 4, m = i & 15;
        sVt[d * P20 + m] = veff[((size_t)b * M + m) * D + d];
    }
    __syncthreads();

    const int ln    = lane & 15;
    const int off   = (lane < 16) ? 0 : 2;   // A/B fragment K-pair select
    const int half8 = (lane < 16) ? 0 : 8;   // C fragment row offset

    // ---- 1) q_proj = Q_tile(16xD) x W_read_key^T(Dx16), 40 x K=4 WMMA ----
    v8f c0 = {};
    const float* qrow = query + ((size_t)b * S + s0 + ln) * D;
    #pragma unroll
    for (int kk = 0; kk < D; kk += 4) {
        v2f a  = *(const v2f*)(qrow + kk + off);
        v2f bb = *(const v2f*)(&sW[ln * WPAD + kk + off]);
        c0 = __builtin_amdgcn_wmma_f32_16x16x4_f32(false, a, false, bb,
                                                   (short)0, c0, false, false);
    }
    #pragma unroll
    for (int v = 0; v < 8; ++v)
        sQp[w][(v + half8) * P20 + ln] = c0[v];

    // ---- 2) scores = q_proj(16x16) x mem_proj^T(16x16), scaled 1/sqrt(R) ----
    v8f c1 = {};
    #pragma unroll
    for (int kk = 0; kk < 16; kk += 4) {
        v2f a  = *(const v2f*)(&sQp[w][ln * P20 + kk + off]);
        v2f bb = *(const v2f*)(&sMem[ln * P20 + kk + off]);
        c1 = __builtin_amdgcn_wmma_f32_16x16x4_f32(false, a, false, bb,
                                                   (short)0, c1, false, false);
    }
    #pragma unroll
    for (int v = 0; v < 8; ++v)
        sAt[w][(v + half8) * P20 + ln] = c1[v] * 0.25f;   // 1/sqrt(16)

    // ---- 3) softmax over m (16 cols); lanes 0..15 own one row each ----
    if (lane < 16) {
        float* row = &sAt[w][lane * P20];
        float mx = row[0];
        #pragma unroll
        for (int j = 1; j < 16; ++j) mx = fmaxf(mx, row[j]);
        float sum = 0.f;
        #pragma unroll
        for (int j = 0; j < 16; ++j) { float e = __expf(row[j] - mx); row[j] = e; sum += e; }
        float inv = 1.0f / sum;
        #pragma unroll
        for (int j = 0; j < 16; ++j) row[j] *= inv;
    }

    // ---- 4) read_out tile = attn(16x16) x V_eff(16xD), 10 x (4 x K=4) WMMA ----
    float* orow = read_out + ((size_t)b * S + s0) * D;
    #pragma unroll
    for (int d0 = 0; d0 < D; d0 += 16) {
        v8f c2 = {};
        #pragma unroll
        for (int kk = 0; kk < 16; kk += 4) {
            v2f a  = *(const v2f*)(&sAt[w][ln * P20 + kk + off]);
            v2f bb = *(const v2f*)(&sVt[(d0 + ln) * P20 + kk + off]);
            c2 = __builtin_amdgcn_wmma_f32_16x16x4_f32(false, a, false, bb,
                                                       (short)0, c2, false, false);
        }
        #pragma unroll
        for (int v = 0; v < 8; ++v)
            orow[(size_t)(v + half8) * D + d0 + ln] = c2[v];
    }
}

// ---------------------------------------------------------------------------
extern "C" void kernel_launch(void* const* d_in, const int* in_sizes, int n_in,
                              void* d_out, int out_size, void* d_ws, size_t ws_size,
                              hipStream_t stream)
{
    const float* query  = (const float*)d_in[0];
    const float* interf = (const float*)d_in[1];
    const float* memory = (const float*)d_in[2];
    const float* Wrk    = (const float*)d_in[3];  // W_read_key   (R,D)
    const float* Wrv    = (const float*)d_in[4];  // W_read_value (D,R)
    const float* Wwk    = (const float*)d_in[5];  // W_write_key  (R,D)
    const float* Wwe    = (const float*)d_in[6];  // W_write_erase(D,R)
    const float* Wwa    = (const float*)d_in[7];  // W_write_add  (D,R)

    float* read_out = (float*)d_out;
    float* new_mem  = read_out + (size_t)B * S * D;

    float* partial = (float*)d_ws;                     // B*CH*D floats
    float* veff    = partial + (size_t)B * CH * D;     // B*M*D floats

    reduce_kernel<<<B * CH, D, 0, stream>>>(interf, partial);
    update_kernel<<<B, D, 0, stream>>>(partial, memory, Wwk, Wwe, Wwa, Wrv,
                                       veff, new_mem);
    attn_kernel<<<B * (S / 128), 256, 0, stream>>>(query, memory, Wrk, veff,
                                                   read_out);
}